// AugmentationLayer_86517821216236
// MI455X (gfx1250) — compile-verified
//
#include <hip/hip_runtime.h>
#include <math.h>

typedef __attribute__((ext_vector_type(2))) float v2f;
typedef __attribute__((ext_vector_type(8))) float v8f;

#define BATCH 64
#define KTR   8
#define FEAT  24
#define DIN   32
#define HID   64
#define LSEQ  2048
#define CCH   64

#define XAUG_ELEMS (BATCH * LSEQ * CCH)          // 8388608
#define PROB_OFF   (XAUG_ELEMS)                  // 8388608
#define INT_OFF    (PROB_OFF + BATCH * KTR)      // 8389120
#define IDX_OFF    (INT_OFF + BATCH * KTR)       // 8389632

static __device__ __forceinline__ v8f wmma_f32(v2f a, v2f b, v8f c) {
    return __builtin_amdgcn_wmma_f32_16x16x4_f32(
        /*neg_a=*/false, a, /*neg_b=*/false, b,
        /*c_mod=*/(short)0, c, /*reuse_a=*/false, /*reuse_b=*/false);
}

// One workgroup, 16 waves (512 threads). Computes both MLPs with fp32 WMMA,
// then softmax / softplus / gumbel-argmax per batch row.
__global__ void __launch_bounds__(512)
mlp_select_kernel(const float* __restrict__ prev_prob,
                  const float* __restrict__ features,
                  const float* __restrict__ gnoise,
                  const float* __restrict__ log_tau,
                  const float* __restrict__ pW1, const float* __restrict__ pb1,
                  const float* __restrict__ pW2, const float* __restrict__ pb2,
                  const float* __restrict__ pW3, const float* __restrict__ pb3,
                  const float* __restrict__ iW1, const float* __restrict__ ib1,
                  const float* __restrict__ iW2, const float* __restrict__ ib2,
                  const float* __restrict__ iW3, const float* __restrict__ ib3,
                  float* __restrict__ prob_out,
                  float* __restrict__ intens_out,
                  float* __restrict__ selidx_out,
                  int*   __restrict__ idx_ws,
                  float* __restrict__ intens_ws) {
    __shared__ float Xs[BATCH][DIN + 4];   // mlp_in, padded
    __shared__ float h1p[BATCH][HID + 4];
    __shared__ float h1i[BATCH][HID + 4];
    __shared__ float h2p[BATCH][HID + 4];
    __shared__ float h2i[BATCH][HID + 4];
    __shared__ float lgp[BATCH][KTR];
    __shared__ float lgi[BATCH][KTR];

    const int tid  = threadIdx.x;
    const int wave = tid >> 5;
    const int lane = tid & 31;
    const int half = lane >> 4;          // which 16-lane half
    const int lm   = lane & 15;

    // ---- stage mlp_in = concat(prev_prob, features) into LDS ----
    for (int e = tid; e < BATCH * DIN; e += blockDim.x) {
        int b = e >> 5, j = e & 31;
        Xs[b][j] = (j < KTR) ? prev_prob[b * KTR + j] : features[b * FEAT + (j - KTR)];
    }
    __syncthreads();

    const int mi = wave >> 2, ni = wave & 3;
    const int m = mi * 16 + lm;          // A-fragment row (lanes 0-15 and 16-31 both hold M=0..15)
    const int n = ni * 16 + lm;          // B/C/D column

    // ---- layer 1: (64x32)@(32x64), K=32 -> 8 wmma steps ----
    {
        v8f accP = {}, accI = {};
        for (int k = 0; k < DIN; k += 4) {
            int k0 = k + 2 * half;
            v2f a;  a.x = Xs[m][k0];          a.y = Xs[m][k0 + 1];
            v2f bp; bp.x = pW1[k0 * HID + n]; bp.y = pW1[(k0 + 1) * HID + n];
            v2f bi; bi.x = iW1[k0 * HID + n]; bi.y = iW1[(k0 + 1) * HID + n];
            accP = wmma_f32(a, bp, accP);
            accI = wmma_f32(a, bi, accI);
        }
        float bbp = pb1[n], bbi = ib1[n];
        #pragma unroll
        for (int r = 0; r < 8; ++r) {
            int mr = mi * 16 + r + 8 * half;   // C/D layout: VGPR r -> M=r (lo half) / r+8 (hi half)
            float vp = accP[r] + bbp; h1p[mr][n] = vp > 0.f ? vp : 0.f;
            float vi = accI[r] + bbi; h1i[mr][n] = vi > 0.f ? vi : 0.f;
        }
    }
    __syncthreads();

    // ---- layer 2: (64x64)@(64x64), K=64 -> 16 wmma steps ----
    {
        v8f accP = {}, accI = {};
        for (int k = 0; k < HID; k += 4) {
            int k0 = k + 2 * half;
            v2f ap; ap.x = h1p[m][k0];        ap.y = h1p[m][k0 + 1];
            v2f ai; ai.x = h1i[m][k0];        ai.y = h1i[m][k0 + 1];
            v2f bp; bp.x = pW2[k0 * HID + n]; bp.y = pW2[(k0 + 1) * HID + n];
            v2f bi; bi.x = iW2[k0 * HID + n]; bi.y = iW2[(k0 + 1) * HID + n];
            accP = wmma_f32(ap, bp, accP);
            accI = wmma_f32(ai, bi, accI);
        }
        float bbp = pb2[n], bbi = ib2[n];
        #pragma unroll
        for (int r = 0; r < 8; ++r) {
            int mr = mi * 16 + r + 8 * half;
            float vp = accP[r] + bbp; h2p[mr][n] = vp > 0.f ? vp : 0.f;
            float vi = accI[r] + bbi; h2i[mr][n] = vi > 0.f ? vi : 0.f;
        }
    }
    __syncthreads();

    // ---- layer 3: (64x64)@(64x8), zero-padded to N=16. waves 0-3: prob, 4-7: intensity ----
    if (wave < 8) {
        const int isI = wave >> 2;
        const int mi2 = wave & 3;
        const float* W3 = isI ? iW3 : pW3;
        const float* b3 = isI ? ib3 : pb3;
        float (*h2)[HID + 4] = isI ? h2i : h2p;
        v8f acc = {};
        for (int k = 0; k < HID; k += 4) {
            int k0 = k + 2 * half;
            int mr = mi2 * 16 + lm;
            v2f a; a.x = h2[mr][k0]; a.y = h2[mr][k0 + 1];
            v2f b;
            b.x = (lm < KTR) ? W3[k0 * KTR + lm]       : 0.f;
            b.y = (lm < KTR) ? W3[(k0 + 1) * KTR + lm] : 0.f;
            acc = wmma_f32(a, b, acc);
        }
        if (lm < KTR) {
            float bias = b3[lm];
            #pragma unroll
            for (int r = 0; r < 8; ++r) {
                int mr = mi2 * 16 + r + 8 * half;
                float v = acc[r] + bias;
                if (isI) lgi[mr][lm] = v; else lgp[mr][lm] = v;
            }
        }
    }
    __syncthreads();

    // ---- per-row epilogue: softmax, softplus, gumbel-argmax ----
    if (tid < BATCH) {
        const int b = tid;
        float tau = expf(log_tau[0]);
        tau = fminf(fmaxf(tau, 0.01f), 10.0f);

        float pl[KTR], il[KTR];
        #pragma unroll
        for (int j = 0; j < KTR; ++j) { pl[j] = lgp[b][j]; il[j] = lgi[b][j]; }

        // prob = softmax(prob_logits)
        float mx = pl[0];
        #pragma unroll
        for (int j = 1; j < KTR; ++j) mx = fmaxf(mx, pl[j]);
        float ex[KTR], s = 0.f;
        #pragma unroll
        for (int j = 0; j < KTR; ++j) { ex[j] = expf(pl[j] - mx); s += ex[j]; }
        float inv = 1.0f / s;
        #pragma unroll
        for (int j = 0; j < KTR; ++j) prob_out[b * KTR + j] = ex[j] * inv;

        // intensity = softplus(intensity_logits), numerically stable like jax.nn.softplus
        #pragma unroll
        for (int j = 0; j < KTR; ++j) {
            float v = il[j];
            float sp = fmaxf(v, 0.f) + log1pf(expf(-fabsf(v)));
            intens_out[b * KTR + j] = sp;
            intens_ws[b * KTR + j]  = sp;
        }

        // argmax of softmax((pl+g)/tau) == argmax of (pl+g)/tau; first-max tie rule
        float it = 1.0f / tau;
        int best = 0;
        float bv = (pl[0] + gnoise[b * KTR + 0]) * it;
        #pragma unroll
        for (int j = 1; j < KTR; ++j) {
            float g = (pl[j] + gnoise[b * KTR + j]) * it;
            if (g > bv) { bv = g; best = j; }
        }
        idx_ws[b] = best;
        selidx_out[b] = (float)best;
    }
}

// Streaming kernel: one selected transform per batch row. float4 along C (C=64),
// so the L-flip (case 7) stays vector-aligned. Branch is wave-uniform.
__global__ void __launch_bounds__(256)
apply_kernel(const float* __restrict__ x,
             const int*   __restrict__ idx_ws,
             const float* __restrict__ intens_ws,
             float* __restrict__ out) {
    const int b   = blockIdx.y;
    const int sel = idx_ws[b];
    const int v   = blockIdx.x * blockDim.x + threadIdx.x;   // float4 index within row, [0, 32768)

    const size_t base = (size_t)b * (LSEQ * CCH);
    const float4* xin = (const float4*)(x + base);
    float4*       xo  = (float4*)(out + base);

    float t = 0.f;
    if (sel >= 1 && sel <= 6) t = intens_ws[b * KTR + sel];

    float4 a, r;
    switch (sel) {
    case 0:
        r = xin[v];
        break;
    case 1: {
        float s = 1.0f + t; a = xin[v];
        r.x = a.x * s; r.y = a.y * s; r.z = a.z * s; r.w = a.w * s;
        break; }
    case 2: {
        a = xin[v];
        r.x = a.x + t; r.y = a.y + t; r.z = a.z + t; r.w = a.w + t;
        break; }
    case 3: {
        float s = 1.0f - t; a = xin[v];
        r.x = a.x * s; r.y = a.y * s; r.z = a.z * s; r.w = a.w * s;
        break; }
    case 4: {
        float s = 1.0f + t; a = xin[v];
        r.x = tanhf(a.x * s); r.y = tanhf(a.y * s);
        r.z = tanhf(a.z * s); r.w = tanhf(a.w * s);
        break; }
    case 5: {
        float s = expf(-t); a = xin[v];
        r.x = a.x * s; r.y = a.y * s; r.z = a.z * s; r.w = a.w * s;
        break; }
    case 6: {
        a = xin[v];
        r.x = a.x + t * sinf(a.x); r.y = a.y + t * sinf(a.y);
        r.z = a.z + t * sinf(a.z); r.w = a.w + t * sinf(a.w);
        break; }
    default: {  // 7: time reversal along L; 16 float4 per L-row
        int l = v >> 4, cv = v & 15;
        r = xin[(size_t)(LSEQ - 1 - l) * 16 + cv];
        break; }
    }
    xo[v] = r;
}

extern "C" void kernel_launch(void* const* d_in, const int* in_sizes, int n_in,
                              void* d_out, int out_size, void* d_ws, size_t ws_size,
                              hipStream_t stream) {
    (void)in_sizes; (void)n_in; (void)out_size; (void)ws_size;
    const float* x         = (const float*)d_in[0];
    const float* prev_prob = (const float*)d_in[1];
    const float* features  = (const float*)d_in[2];
    const float* gnoise    = (const float*)d_in[3];
    const float* log_tau   = (const float*)d_in[4];
    const float* pW1 = (const float*)d_in[5];
    const float* pb1 = (const float*)d_in[6];
    const float* pW2 = (const float*)d_in[7];
    const float* pb2 = (const float*)d_in[8];
    const float* pW3 = (const float*)d_in[9];
    const float* pb3 = (const float*)d_in[10];
    const float* iW1 = (const float*)d_in[11];
    const float* ib1 = (const float*)d_in[12];
    const float* iW2 = (const float*)d_in[13];
    const float* ib2 = (const float*)d_in[14];
    const float* iW3 = (const float*)d_in[15];
    const float* ib3 = (const float*)d_in[16];

    float* out = (float*)d_out;
    int*   idx_ws    = (int*)d_ws;
    float* intens_ws = (float*)((char*)d_ws + 256);

    mlp_select_kernel<<<dim3(1), dim3(512), 0, stream>>>(
        prev_prob, features, gnoise, log_tau,
        pW1, pb1, pW2, pb2, pW3, pb3,
        iW1, ib1, iW2, ib2, iW3, ib3,
        out + PROB_OFF, out + INT_OFF, out + IDX_OFF,
        idx_ws, intens_ws);

    apply_kernel<<<dim3((LSEQ * CCH) / 4 / 256, BATCH), dim3(256), 0, stream>>>(
        x, idx_ws, intens_ws, out);
}